// Model_23880018165989
// MI455X (gfx1250) — compile-verified
//
#include <hip/hip_runtime.h>
#include <hip/hip_bf16.h>

// ---------------- problem constants ----------------
#define BB     4
#define SS     1024
#define DM     192
#define NH     16
#define DH     6
#define INNER  96
#define SCALE  0.40824829046386302f   // 1/sqrt(6)
#define RHT    2064                   // 2*S + 16 pad rows
#define JT     64                     // key tiles per row block

// workspace layout (float offsets)
#define QW_OFF 0u
#define QR_OFF 524288u                // 4*16*1024*8
#define KP_OFF 1048576u
#define VP_OFF 1572864u               // Vp padded to 16 cols -> 1048576 floats
#define RH_OFF 2621440u               // 16*2064*8 = 264192
#define AV_OFF 2885632u               // 4*1024*96 = 393216
#define WS_FLOATS 3278848u

typedef __attribute__((ext_vector_type(2))) float v2f;
typedef __attribute__((ext_vector_type(4))) float v4f;
typedef __attribute__((ext_vector_type(8))) float v8f;
typedef int v4i_vs __attribute__((vector_size(16)));   // matches builtin param type

static __device__ inline v8f wmma4(v2f a, v2f b, v8f c) {
  // D = A(16x4, f32) * B(4x16, f32) + C(16x16, f32)
  return __builtin_amdgcn_wmma_f32_16x16x4_f32(false, a, false, b, (short)0, c,
                                               false, false);
}

// ---- gfx1250 async global->LDS copy (ASYNCcnt), with safe fallback ----
#if defined(__has_builtin)
#if __has_builtin(__builtin_amdgcn_global_load_async_to_lds_b128)
#define HAVE_ASYNC_LDS 1
#endif
#endif

static __device__ inline void g2l_b128(const float* g, float* lds) {
#ifdef HAVE_ASYNC_LDS
  __builtin_amdgcn_global_load_async_to_lds_b128(
      (__attribute__((address_space(1))) v4i_vs*)(uintptr_t)g,
      (__attribute__((address_space(3))) v4i_vs*)(unsigned)(uintptr_t)lds,
      0, 0);
#else
  *(v4f*)lds = *(const v4f*)g;   // synchronous fallback
#endif
}

template <int N>
static __device__ inline void async_wait_le() {
#ifdef HAVE_ASYNC_LDS
#if __has_builtin(__builtin_amdgcn_s_wait_asynccnt)
  __builtin_amdgcn_s_wait_asynccnt(N);
#else
  asm volatile("s_wait_asynccnt %0" ::"i"(N) : "memory");
#endif
#endif
  asm volatile("" ::: "memory");   // keep LDS reads below the wait
}

// ---------------- 0) zero the padded workspace ----------------
__global__ void zero_ws_kernel(float* ws, unsigned n) {
  unsigned i = blockIdx.x * blockDim.x + threadIdx.x;
  unsigned stride = gridDim.x * blockDim.x;
  for (; i < n; i += stride) ws[i] = 0.0f;
}

// ---------------- 1) fused QKV projection (WMMA GEMM) ----------------
// grid (256 Mtiles, 6 Ntiles), block 32 (one wave per 16x16 output tile)
__global__ void proj_kernel(const float* __restrict__ x,
                            const float* __restrict__ Wq,
                            const float* __restrict__ Wk,
                            const float* __restrict__ bkv,
                            const float* __restrict__ Wv,
                            const float* __restrict__ bvv,
                            const float* __restrict__ rwb,
                            const float* __restrict__ rrb,
                            float* __restrict__ ws) {
  const int l = threadIdx.x, lh = l >> 4, lm = l & 15;
  const int row0 = blockIdx.x * 16, n0 = blockIdx.y * 16;
  v8f cq = {}, ck = {}, cv = {};
  for (int kk = 0; kk < 48; ++kk) {
    const int krow = kk * 4 + 2 * lh;             // K index pair base for this lane
    v2f a = *(const v2f*)(x + (row0 + lm) * DM + krow);
    v2f bq = { Wq[krow * INNER + n0 + lm], Wq[(krow + 1) * INNER + n0 + lm] };
    v2f bk = { Wk[krow * INNER + n0 + lm], Wk[(krow + 1) * INNER + n0 + lm] };
    v2f bv = { Wv[krow * INNER + n0 + lm], Wv[(krow + 1) * INNER + n0 + lm] };
    cq = wmma4(a, bq, cq);
    ck = wmma4(a, bk, ck);
    cv = wmma4(a, bv, cv);
  }
  const int col  = n0 + lm;
  const int head = col / DH;
  const int h    = col - head * DH;
  float* QW = ws + QW_OFF; float* QR = ws + QR_OFF;
  float* KP = ws + KP_OFF; float* VP = ws + VP_OFF;
  for (int v = 0; v < 8; ++v) {
    const int row = row0 + v + 8 * lh;
    const int b = row >> 10, il = row & 1023;
    const unsigned base8  = ((unsigned)(b * NH + head) * SS + il) * 8u + h;
    const unsigned base16 = ((unsigned)(b * NH + head) * SS + il) * 16u + h;
    QW[base8]  = SCALE * (cq[v] + rwb[col]);
    QR[base8]  = SCALE * (cq[v] + rrb[col]);
    KP[base8]  = ck[v] + bkv[col];
    VP[base16] = cv[v] + bvv[col];
  }
}

// ---------------- 2) r_head = rel_pos_embeds @ r_kernel ----------------
__global__ void rh_kernel(const float* __restrict__ r_kernel, float* __restrict__ ws) {
  __shared__ float rbuf[DM];
  const int t = blockIdx.x, tid = threadIdx.x;
  const float rel = (float)(SS - t);
  const int f = (tid < 96) ? tid : tid - 96;
  const float invf = __expf((float)f * (-9.210340371976184f / 96.0f)); // 10000^{-f/96}
  const float ang = rel * invf;
  rbuf[tid] = (tid < 96) ? sinf(ang) : cosf(ang);
  __syncthreads();
  if (tid < INNER) {
    float acc = 0.0f;
    for (int d = 0; d < DM; ++d) acc += rbuf[d] * r_kernel[d * INNER + tid];
    const int n = tid / DH, h = tid - n * DH;
    ws[RH_OFF + ((unsigned)n * RHT + t) * 8u + h] = acc;
  }
}

// ---------------- 3) fused flash attention with relative shift ----------------
// grid (S/16, NH, B), block 32 (one wave). Double-buffered async K/RH/V tiles.
__global__ void attn_kernel(const float* __restrict__ mask0, float* __restrict__ ws) {
  __shared__ __align__(16) float kbuf[2][16][8];    // key tile
  __shared__ __align__(16) float rbuf[2][32][8];    // positional band
  __shared__ __align__(16) float vbuf[2][16][16];   // value tile
  __shared__ float pbuf[16][32];
  __shared__ float sbuf[16][16];
  __shared__ float fbuf[16];
  const int l = threadIdx.x, lh = l >> 4, lm = l & 15;
  const int i0 = blockIdx.x * 16, head = blockIdx.y, b = blockIdx.z;
  const unsigned bh = (unsigned)(b * NH + head);

  const float* QW = ws + QW_OFF + (bh * SS) * 8u;
  const float* QR = ws + QR_OFF + (bh * SS) * 8u;
  const float* KP = ws + KP_OFF + (bh * SS) * 8u;
  const float* VP = ws + VP_OFF + (bh * SS) * 16u;
  const float* RH = ws + RH_OFF + (unsigned)head * RHT * 8u;
  float*       AV = ws + AV_OFF;

  // query fragments (A layout: m = lane%16, k = 2*(lane/16)+v), two K-steps
  v2f aw0 = *(const v2f*)(QW + (i0 + lm) * 8 + 2 * lh);
  v2f aw1 = *(const v2f*)(QW + (i0 + lm) * 8 + 4 + 2 * lh);
  v2f ar0 = *(const v2f*)(QR + (i0 + lm) * 8 + 2 * lh);
  v2f ar1 = *(const v2f*)(QR + (i0 + lm) * 8 + 4 + 2 * lh);

  v8f acc = {};
  float m_r = -INFINITY, l_r = 0.0f;     // owned by lanes 0..15 (row = lane)

  // issue the 5 async B128 copies for key-tile t into buffer t&1
  auto issue_tile = [&](int t) {
    const int j0 = t * 16;
    const int buf = t & 1;
    const int tb = SS - i0 + j0 - 15;
    g2l_b128(KP + j0 * 8 + l * 4,        &kbuf[buf][0][0] + l * 4);
    g2l_b128(RH + tb * 8 + l * 4,        &rbuf[buf][0][0] + l * 4);
    g2l_b128(RH + tb * 8 + 128 + l * 4,  &rbuf[buf][0][0] + 128 + l * 4);
    g2l_b128(VP + j0 * 16 + l * 4,       &vbuf[buf][0][0] + l * 4);
    g2l_b128(VP + j0 * 16 + 128 + l * 4, &vbuf[buf][0][0] + 128 + l * 4);
  };

  auto compute_tile = [&](int t) {
    const int buf = t & 1;
    const int j0 = t * 16;
    // ---- content scores ----
    v8f c = {};
    c = wmma4(aw0, *(const v2f*)&kbuf[buf][lm][2 * lh], c);
    c = wmma4(aw1, *(const v2f*)&kbuf[buf][lm][4 + 2 * lh], c);
    // ---- positional band P = QR @ RH[tb..tb+31]^T ----
    v8f p0 = {}, p1 = {};
    p0 = wmma4(ar0, *(const v2f*)&rbuf[buf][lm][2 * lh], p0);
    p0 = wmma4(ar1, *(const v2f*)&rbuf[buf][lm][4 + 2 * lh], p0);
    p1 = wmma4(ar0, *(const v2f*)&rbuf[buf][16 + lm][2 * lh], p1);
    p1 = wmma4(ar1, *(const v2f*)&rbuf[buf][16 + lm][4 + 2 * lh], p1);
    for (int v = 0; v < 8; ++v) {
      const int di = v + 8 * lh;
      pbuf[di][lm] = p0[v];
      pbuf[di][16 + lm] = p1[v];
    }
    __syncthreads();
    // relative-shift gather + mask -> score tile
    const float madd = -1.0e6f * (1.0f - mask0[b * SS + j0 + lm]);
    for (int v = 0; v < 8; ++v) {
      const int di = v + 8 * lh;
      sbuf[di][lm] = c[v] + pbuf[di][15 - di + lm] + madd;
    }
    __syncthreads();
    // online softmax: lanes 0..15 each own one row
    if (l < 16) {
      float tmax = sbuf[l][0];
      for (int j = 1; j < 16; ++j) tmax = fmaxf(tmax, sbuf[l][j]);
      const float mnew = fmaxf(m_r, tmax);
      const float factor = __expf(m_r - mnew);
      float s = 0.0f;
      for (int j = 0; j < 16; ++j) {
        const float e = __expf(sbuf[l][j] - mnew);
        sbuf[l][j] = e;
        s += e;
      }
      l_r = l_r * factor + s;
      m_r = mnew;
      fbuf[l] = factor;
    }
    __syncthreads();
    // rescale accumulator (C fragment) and add P @ V
    for (int v = 0; v < 8; ++v) acc[v] *= fbuf[v + 8 * lh];
    for (int kk = 0; kk < 4; ++kk) {
      const int kr = 4 * kk + 2 * lh;
      v2f a = { sbuf[lm][kr], sbuf[lm][kr + 1] };
      v2f bv = { vbuf[buf][kr][lm], vbuf[buf][kr + 1][lm] };
      acc = wmma4(a, bv, acc);
    }
    __syncthreads();   // protect pbuf/sbuf before next tile
  };

  issue_tile(0);
  for (int t = 0; t < JT - 1; ++t) {
    issue_tile(t + 1);      // prefetch next tile behind this tile's compute
    async_wait_le<5>();     // in-order completion => tile t resident
    compute_tile(t);
  }
  async_wait_le<0>();
  compute_tile(JT - 1);

  // normalize and write attn_vec (only h < 6 columns are real)
  if (l < 16) fbuf[l] = 1.0f / l_r;
  __syncthreads();
  for (int v = 0; v < 8; ++v) {
    const int di = v + 8 * lh;
    if (lm < DH)
      AV[((unsigned)(b * SS + i0 + di)) * INNER + head * DH + lm] = acc[v] * fbuf[di];
  }
}

// ---------------- 4) output projection + residual + LayerNorm ----------------
__global__ void out_ln_kernel(const float* __restrict__ x,
                              const float* __restrict__ Wo,
                              const float* __restrict__ bo,
                              const float* __restrict__ gamma,
                              const float* __restrict__ beta,
                              const float* __restrict__ ws,
                              float* __restrict__ out) {
  __shared__ float avr[INNER];
  __shared__ float red[256];
  const int row = blockIdx.x, t = threadIdx.x;
  if (t < INNER) avr[t] = ws[AV_OFF + (unsigned)row * INNER + t];
  __syncthreads();
  float acc = bo[t];
  for (int d = 0; d < INNER; ++d) acc += avr[d] * Wo[d * DM + t];
  const float h = x[row * DM + t] + acc;
  red[t] = h;
  if (t < 64) red[DM + t] = 0.0f;
  __syncthreads();
  for (int s = 128; s > 0; s >>= 1) {
    if (t < s) red[t] += red[t + s];
    __syncthreads();
  }
  const float mu = red[0] * (1.0f / DM);
  __syncthreads();
  const float d = h - mu;
  red[t] = d * d;
  if (t < 64) red[DM + t] = 0.0f;
  __syncthreads();
  for (int s = 128; s > 0; s >>= 1) {
    if (t < s) red[t] += red[t + s];
    __syncthreads();
  }
  const float var = red[0] * (1.0f / DM);
  out[row * DM + t] = d * rsqrtf(var + 1e-9f) * gamma[t] + beta[t];
}

// ---------------- launcher ----------------
extern "C" void kernel_launch(void* const* d_in, const int* in_sizes, int n_in,
                              void* d_out, int out_size, void* d_ws, size_t ws_size,
                              hipStream_t stream) {
  const float* x     = (const float*)d_in[0];
  const float* mask0 = (const float*)d_in[1];
  const float* Wq    = (const float*)d_in[2];
  const float* Wk    = (const float*)d_in[3];
  const float* bk    = (const float*)d_in[4];
  const float* Wv    = (const float*)d_in[5];
  const float* bv    = (const float*)d_in[6];
  const float* Wo    = (const float*)d_in[7];
  const float* bo    = (const float*)d_in[8];
  const float* rwb   = (const float*)d_in[9];
  const float* rrb   = (const float*)d_in[10];
  const float* rker  = (const float*)d_in[11];
  const float* gamma = (const float*)d_in[12];
  const float* beta  = (const float*)d_in[13];
  float* out = (float*)d_out;
  float* ws  = (float*)d_ws;

  (void)in_sizes; (void)n_in; (void)out_size; (void)ws_size;

  zero_ws_kernel<<<1024, 256, 0, stream>>>(ws, AV_OFF);
  proj_kernel<<<dim3(BB * SS / 16, INNER / 16), 32, 0, stream>>>(
      x, Wq, Wk, bk, Wv, bv, rwb, rrb, ws);
  rh_kernel<<<2 * SS, DM, 0, stream>>>(rker, ws);
  attn_kernel<<<dim3(SS / 16, NH, BB), 32, 0, stream>>>(mask0, ws);
  out_ln_kernel<<<BB * SS, DM, 0, stream>>>(x, Wo, bo, gamma, beta, ws, out);
}